// loss_kernelMatching_perpixel_14499809591888
// MI455X (gfx1250) — compile-verified
//
#include <hip/hip_runtime.h>
#include <math.h>
#include <stddef.h>

// Problem constants (fixed by reference)
#define N   1681
#define BS  32

typedef __attribute__((ext_vector_type(2))) float v2f;
typedef __attribute__((ext_vector_type(8))) float v8f;

// -----------------------------------------------------------------------------
// Pass 1: per-row L2 norms of pred & target -> w[b,j] = 1/(||p_row||*||t_row||)
// One wave32 per (b,row). Contiguous coalesced reads (lane-strided by 32).
// -----------------------------------------------------------------------------
__global__ __launch_bounds__(256) void rownorm_kernel(
    const float* __restrict__ pred,
    const float* __restrict__ target,
    float* __restrict__ w)
{
    const int gw   = (int)((blockIdx.x * blockDim.x + threadIdx.x) >> 5); // wave id
    const int lane = (int)(threadIdx.x & 31);
    const int b = gw / N;
    const int r = gw - b * N;

    const float* __restrict__ pr = pred   + ((size_t)b * N + r) * (size_t)N;
    const float* __restrict__ tr = target + ((size_t)b * N + r) * (size_t)N;

    float sp = 0.0f, st = 0.0f;
    #pragma unroll 4
    for (int k = lane; k < N; k += 32) {
        float pv = pr[k];
        float tv = tr[k];
        sp = fmaf(pv, pv, sp);
        st = fmaf(tv, tv, st);
    }
    // wave32 butterfly reduction
    for (int off = 16; off > 0; off >>= 1) {
        sp += __shfl_xor(sp, off, 32);
        st += __shfl_xor(st, off, 32);
    }
    if (lane == 0) {
        w[gw] = 1.0f / (sqrtf(sp) * sqrtf(st));
    }
}

// -----------------------------------------------------------------------------
// Pass 2: column sums of pred*target via V_WMMA_F32_16X16X4_F32 (A = ones).
// B layout (4x16 f32, 2 VGPRs/lane):
//   VGPR0: lane L -> B[L>>4][L&15]      (k = 0/1, n = 0..15)
//   VGPR1: lane L -> B[2+(L>>4)][L&15]  (k = 2/3)
// A = ones => D[m][n] = sum_k B[k][n], accumulated exactly in fp32.
//
// Branch-free hot loop:
//  - Rows: quads 0..419 cover rows 0..1679 (always in-bounds). Row 1680 is
//    added as a scalar term in the reduction stage.
//  - Cols: j0 = 32*tile+n <= 1679 is always valid; only j1 = j0+16 can
//    overflow (tile 52) -> clamp address once, multiply by 0/1 mask.
//  - Addressing: SGPR base + 32-bit per-lane element index (scale_offset
//    form); one v_add_nc_u32 per index per iteration, +2-row constant folds
//    into the 24-bit load immediate.  8 unconditional loads / iteration.
// -----------------------------------------------------------------------------
__global__ __launch_bounds__(256) void coldot_wmma_kernel(
    const float* __restrict__ pred,
    const float* __restrict__ target,
    const float* __restrict__ w,
    float* __restrict__ partial)
{
    const int b    = (int)blockIdx.y;          // batch
    const int tile = (int)blockIdx.x;          // 32-column tile, 0..52
    const int wave = (int)(threadIdx.x >> 5);  // 0..7
    const int lane = (int)(threadIdx.x & 31);
    const int kq   = lane >> 4;                // 0/1: k row within B VGPR
    const int n    = lane & 15;                // column within 16-wide half-tile

    const int   j0  = tile * 32 + n;           // always < N
    const int   j1  = j0 + 16;
    const int   jc1 = (j1 < N) ? j1 : (N - 1); // clamped (tile 52 only)
    const float mj1 = (j1 < N) ? 1.0f : 0.0f;  // 0/1 column mask

    const size_t base = (size_t)b * (size_t)N * (size_t)N;
    const float* __restrict__ P = pred + base;
    const float* __restrict__ T = target + base;

    const unsigned row0  = (unsigned)(4 * wave + kq) * (unsigned)N;
    const unsigned TWO   = 2u  * (unsigned)N;  // +2 rows (13448 B immediate)
    const unsigned STEPI = 32u * (unsigned)N;  // 8 waves * 4 rows

    unsigned idx0 = row0 + (unsigned)j0;       // shared index for P and T
    unsigned idx1 = row0 + (unsigned)jc1;

    v2f a; a[0] = 1.0f; a[1] = 1.0f;           // A = ones(16x4)
    v8f c0 = {};                               // columns j0
    v8f c1 = {};                               // columns j1

    #pragma unroll 2
    for (int q = wave; q < 420; q += 8) {
        float p00 = P[idx0];        float p01 = P[idx0 + TWO];
        float t00 = T[idx0];        float t01 = T[idx0 + TWO];
        float p10 = P[idx1];        float p11 = P[idx1 + TWO];
        float t10 = T[idx1];        float t11 = T[idx1 + TWO];

        v2f B0; B0[0] = p00 * t00;         B0[1] = p01 * t01;
        v2f B1; B1[0] = p10 * t10 * mj1;   B1[1] = p11 * t11 * mj1;

        c0 = __builtin_amdgcn_wmma_f32_16x16x4_f32(
                 false, a, false, B0, (short)0, c0, false, false);
        c1 = __builtin_amdgcn_wmma_f32_16x16x4_f32(
                 false, a, false, B1, (short)0, c1, false, false);

        idx0 += STEPI;
        idx1 += STEPI;
    }

    // c[0] lanes 0..15 hold (M=0, n) = per-column sums; combine 8 waves in LDS.
    __shared__ float lds[8][32];
    if (lane < 16) {
        lds[wave][n]      = c0[0];
        lds[wave][n + 16] = c1[0];
    }
    __syncthreads();

    if (wave == 0) {
        float s = 0.0f;
        #pragma unroll
        for (int ww = 0; ww < 8; ++ww) s += lds[ww][lane];
        const int j = tile * 32 + lane;
        if (j < N) {
            const size_t oL = (size_t)(N - 1) * (size_t)N;  // row 1680
            s += P[oL + j] * T[oL + j];                     // leftover row
            s *= w[(size_t)b * N + j];
        } else {
            s = 0.0f;
        }
        for (int off = 16; off > 0; off >>= 1) s += __shfl_xor(s, off, 32);
        if (lane == 0) partial[b * gridDim.x + tile] = s;
    }
}

// -----------------------------------------------------------------------------
// Final: deterministic single-block reduction of per-block partials.
// -----------------------------------------------------------------------------
__global__ __launch_bounds__(256) void finalize_kernel(
    const float* __restrict__ partial, float* __restrict__ out, int nparts)
{
    __shared__ float lds[256];
    float s = 0.0f;
    for (int i = (int)threadIdx.x; i < nparts; i += 256) s += partial[i];
    lds[threadIdx.x] = s;
    __syncthreads();
    for (int stride = 128; stride > 0; stride >>= 1) {
        if ((int)threadIdx.x < stride) lds[threadIdx.x] += lds[threadIdx.x + stride];
        __syncthreads();
    }
    if (threadIdx.x == 0) out[0] = -lds[0] / ((float)BS * 1681.0f);
}

// -----------------------------------------------------------------------------
extern "C" void kernel_launch(void* const* d_in, const int* in_sizes, int n_in,
                              void* d_out, int out_size, void* d_ws, size_t ws_size,
                              hipStream_t stream)
{
    const float* pred   = (const float*)d_in[0];
    const float* target = (const float*)d_in[1];
    float* out = (float*)d_out;

    // workspace: w[BS*N] followed by partial[BS*53]
    float* w       = (float*)d_ws;
    float* partial = w + (size_t)BS * N;

    // Pass 1: 32*1681 = 53792 waves -> 6724 blocks of 256 threads (8 waves)
    rownorm_kernel<<<dim3((BS * N) / 8), dim3(256), 0, stream>>>(pred, target, w);

    // Pass 2: 53 column tiles (32 cols each) x 32 batches
    const int ntiles = (N + 31) / 32;   // 53
    coldot_wmma_kernel<<<dim3(ntiles, BS), dim3(256), 0, stream>>>(pred, target, w, partial);

    // Final scalar
    finalize_kernel<<<dim3(1), dim3(256), 0, stream>>>(partial, out, ntiles * BS);
}